// reweight_loss_266287973106
// MI455X (gfx1250) — compile-verified
//
#include <hip/hip_runtime.h>
#include <stdint.h>

#define CDIM   1000
#define NROWS  32768
#define MAIN_BLOCKS 2048
#define THREADS 256
#define WPB (THREADS / 32)      // waves per block (wave32)
#define BUF_FLOATS 1024         // per-buffer LDS floats (row is 1000)

// ---------------------------------------------------------------------------
// Kernel 1: tiled transpose T[c][y] -> Tt[y][c] so the per-row column gather
// becomes a coalesced, L2-resident row stream (T is 4 MB << 192 MB L2).
// ---------------------------------------------------------------------------
__global__ __launch_bounds__(256) void transpose_kernel(
    const float* __restrict__ T, float* __restrict__ Tt) {
  __shared__ float tile[32][33];                 // +1 pad kills bank conflicts
  const int bx = blockIdx.x * 32;
  const int by = blockIdx.y * 32;
  int x = bx + threadIdx.x;
  int y = by + threadIdx.y;
#pragma unroll
  for (int j = 0; j < 32; j += 8) {
    if (x < CDIM && (y + j) < CDIM)
      tile[threadIdx.y + j][threadIdx.x] = T[(size_t)(y + j) * CDIM + x];
  }
  __syncthreads();
  x = by + threadIdx.x;
  y = bx + threadIdx.y;
#pragma unroll
  for (int j = 0; j < 32; j += 8) {
    if (x < CDIM && (y + j) < CDIM)
      Tt[(size_t)(y + j) * CDIM + x] = tile[threadIdx.x][threadIdx.y + j];
  }
}

// ---------------------------------------------------------------------------
// Issue one row (1000 f32 = 8 x b128/lane-group) as gfx1250 async copies
// into this wave's private LDS buffer. Tracked by ASYNCcnt (8 per row).
// k==7 is lane-predicated (lanes 26..31 OOB) but still counts as 1 issue.
// ---------------------------------------------------------------------------
__device__ __forceinline__ void issue_row_async(uint32_t lds_base_bytes,
                                                const float* __restrict__ gp,
                                                int lane) {
#pragma unroll
  for (int k = 0; k < 8; ++k) {
    const int base = k * 128 + lane * 4;
    if (base < CDIM) {
      const uint32_t laddr = lds_base_bytes + (uint32_t)(base * 4);
      const float*   ga    = gp + base;
      asm volatile("global_load_async_to_lds_b128 %0, %1, off"
                   :: "v"(laddr), "v"(ga)
                   : "memory");
    }
  }
}

// ---------------------------------------------------------------------------
// Kernel 2: one wave32 per row, async double-buffered through LDS.
// Single pass over `out` (131 MB -> ~5.6 us HBM floor).
// beta*ce = (e_y / sum_c e_c*T[c,y]) * (log s - (x_y - m));  s cancels in beta.
// ---------------------------------------------------------------------------
__global__ __launch_bounds__(256) void rowloss_kernel(
    const float* __restrict__ outp, const float* __restrict__ Tt,
    const int* __restrict__ target, float* __restrict__ partials) {
  __shared__ __align__(16) float rowbuf[WPB * 2 * BUF_FLOATS];  // 64 KB/block
  __shared__ float wsum[WPB];

  const int lane   = threadIdx.x & 31;
  const int wid    = threadIdx.x >> 5;
  const int gwave  = blockIdx.x * WPB + wid;
  const int nwaves = gridDim.x * WPB;

  float* mybuf = &rowbuf[wid * 2 * BUF_FLOATS];
  const uint32_t lds0 = (uint32_t)(size_t)(void*)mybuf;

  float acc = 0.0f;
  int cur = 0;

  if (gwave < NROWS)  // prologue: start row 0 of this wave's stream
    issue_row_async(lds0, outp + (size_t)gwave * CDIM, lane);

  for (int row = gwave; row < NROWS; row += nwaves) {
    const int  nxt  = row + nwaves;
    const bool have = nxt < NROWS;
    if (have)  // keep next row in flight while we compute on the current one
      issue_row_async(lds0 + (cur ^ 1) * (BUF_FLOATS * 4),
                      outp + (size_t)nxt * CDIM, lane);

    const int y = target[row];                  // wave-uniform scalar load
    const float* ty = Tt + (size_t)y * CDIM;
    __builtin_prefetch(ty + lane * 32, 0, 3);   // global_prefetch_b8: 4 KB Tt row

    // In-order ASYNCcnt completion: <=8 pending => current buffer is done.
    if (have) asm volatile("s_wait_asynccnt 0x8" ::: "memory");
    else      asm volatile("s_wait_asynccnt 0x0" ::: "memory");

    const float* lbuf = mybuf + cur * BUF_FLOATS;

    // Pass 1: row max (ds_load_b128 + wave32 butterfly).
    float m = -3.0e38f;
#pragma unroll
    for (int k = 0; k < 8; ++k) {
      const int base = k * 128 + lane * 4;
      if (base < CDIM) {
        const float4 v = *reinterpret_cast<const float4*>(lbuf + base);
        m = fmaxf(m, fmaxf(fmaxf(v.x, v.y), fmaxf(v.z, v.w)));
      }
    }
#pragma unroll
    for (int off = 16; off > 0; off >>= 1)
      m = fmaxf(m, __shfl_xor(m, off, 32));

    // Pass 2 (fused): e = exp(x-m);  s += e;  d += e * Tt[y,c].
    float s = 0.0f, d = 0.0f;
#pragma unroll
    for (int k = 0; k < 8; ++k) {
      const int base = k * 128 + lane * 4;
      if (base < CDIM) {
        const float4 v = *reinterpret_cast<const float4*>(lbuf + base);
        const float4 t = *reinterpret_cast<const float4*>(ty + base);
        const float e0 = __expf(v.x - m);
        const float e1 = __expf(v.y - m);
        const float e2 = __expf(v.z - m);
        const float e3 = __expf(v.w - m);
        s += (e0 + e1) + (e2 + e3);
        d = fmaf(e0, t.x, d);
        d = fmaf(e1, t.y, d);
        d = fmaf(e2, t.z, d);
        d = fmaf(e3, t.w, d);
      }
    }
#pragma unroll
    for (int off = 16; off > 0; off >>= 1) {
      s += __shfl_xor(s, off, 32);
      d += __shfl_xor(d, off, 32);
    }

    const float xy = lbuf[y];                   // lane-uniform LDS broadcast
    const float ey = __expf(xy - m);
    const float ce = __logf(s) - (xy - m);      // -log_softmax[y]
    acc += (ey / d) * ce;                       // beta * ce (s cancels in beta)

    cur ^= 1;
  }

  if (lane == 0) wsum[wid] = acc;
  __syncthreads();
  if (threadIdx.x == 0) {
    float b = 0.0f;
#pragma unroll
    for (int i = 0; i < WPB; ++i) b += wsum[i];
    partials[blockIdx.x] = b;                   // fixed order -> deterministic
  }
}

// ---------------------------------------------------------------------------
// Kernel 3: deterministic final reduction of per-block partials -> scalar loss.
// ---------------------------------------------------------------------------
__global__ __launch_bounds__(256) void finalize_kernel(
    const float* __restrict__ partials, int nparts, float* __restrict__ out) {
  __shared__ float buf[256];
  float s = 0.0f;
  for (int i = threadIdx.x; i < nparts; i += 256) s += partials[i];
  buf[threadIdx.x] = s;
  __syncthreads();
  for (int st = 128; st > 0; st >>= 1) {
    if (threadIdx.x < st) buf[threadIdx.x] += buf[threadIdx.x + st];
    __syncthreads();
  }
  if (threadIdx.x == 0) out[0] = buf[0] * (1.0f / (float)NROWS);
}

// ---------------------------------------------------------------------------
extern "C" void kernel_launch(void* const* d_in, const int* in_sizes, int n_in,
                              void* d_out, int out_size, void* d_ws, size_t ws_size,
                              hipStream_t stream) {
  const float* outp   = (const float*)d_in[0];   // [N, C] logits
  const float* T      = (const float*)d_in[1];   // [C, C]
  const int*   target = (const int*)d_in[2];     // [N]

  float* Tt       = (float*)d_ws;                                   // 4,000,000 B
  float* partials = (float*)((char*)d_ws + (size_t)CDIM * CDIM * sizeof(float));

  dim3 tb(32, 8);
  dim3 tg((CDIM + 31) / 32, (CDIM + 31) / 32);
  transpose_kernel<<<tg, tb, 0, stream>>>(T, Tt);

  rowloss_kernel<<<MAIN_BLOCKS, THREADS, 0, stream>>>(outp, Tt, target, partials);

  finalize_kernel<<<1, 256, 0, stream>>>(partials, MAIN_BLOCKS, (float*)d_out);
}